// CascadedBilateralSSM_19894288515301
// MI455X (gfx1250) — compile-verified
//
#include <hip/hip_runtime.h>
#include <math.h>

// ---------------- problem constants (from setup_inputs) ----------------
#define BB   8
#define CIN  64
#define CC   128
#define HH   64
#define TT   512
#define MID  8
#define SD   32
#define EPSV 1e-5f

static const size_t HT   = (size_t)HH * TT;        // 32768
static const size_t NBIG = (size_t)BB * CC * HT;   // 33,554,432 elems (128 MB fp32)

typedef __attribute__((ext_vector_type(2))) float v2f;
typedef __attribute__((ext_vector_type(8))) float v8f;
typedef __attribute__((ext_vector_type(4))) int   v4i;

// CDNA5 async global->LDS path (ASYNCcnt-tracked), guarded for toolchain drift.
#if defined(__has_builtin)
#if __has_builtin(__builtin_amdgcn_global_load_async_to_lds_b128) && \
    __has_builtin(__builtin_amdgcn_s_wait_asynccnt)
#define HAVE_ASYNC_LDS 1
#endif
#endif
#ifndef HAVE_ASYNC_LDS
#define HAVE_ASYNC_LDS 0
#endif

__device__ __forceinline__ float sigmoidf_(float x) { return 1.0f / (1.0f + __expf(-x)); }
__device__ __forceinline__ float geluf_(float x) {
  return 0.5f * x * (1.0f + erff(x * 0.70710678118654752f)); // exact GELU
}

// ---------------------------------------------------------------------------
// Pointwise (1x1) conv as WMMA f32 GEMM.  D[m=out_chan, n=t] = sum_k W[m,k]*In[k,t]
// Block = 256 threads (8 waves) = one (b,h,t0) tile x all 128 output channels.
// The 16-t x Cin activation tile is staged into LDS once per block (async
// global->LDS when available), removing the 8x redundant VMEM reads.
// Optional fused 1x3 depthwise conv is applied at staging time (branchless
// clamped loads), so the WMMA inner loop is uniform and divergence-free.
//   epi: 0 = none, 1 = BN+ReLU, 2 = BN+sigmoid
// ---------------------------------------------------------------------------
template <int KDIM>
__global__ void pw_gemm_kernel(const float* __restrict__ in, const float* __restrict__ w,
                               const float* __restrict__ bn, const float* __restrict__ dw3,
                               float* __restrict__ out, int epi)
{
  __shared__ float As[KDIM * 16];              // [k][t] tile, <= 8 KB

  const int tid  = threadIdx.x;
  const int lane = tid & 31;
  const int wv   = tid >> 5;                   // out-channel tile (0..7)
  const int tpr  = TT / 16;                    // 32 t-tiles per row
  const int mt   = blockIdx.x;                 // (b,h,t0) tile
  const int b    = mt / (HH * tpr);
  const int rem  = mt - b * HH * tpr;
  const int h    = rem / tpr;
  const int t0   = (rem - h * tpr) * 16;

  const float* gbase = in + ((size_t)b * KDIM) * HT + (size_t)h * TT + t0;

  // -------- stage activation tile into LDS --------
  if (dw3 == nullptr) {
#if HAVE_ASYNC_LDS
    for (int i = tid; i < KDIM * 4; i += 256) {        // b128 = 4 floats each
      const int k  = i >> 2;
      const int tq = (i & 3) * 4;
      __builtin_amdgcn_global_load_async_to_lds_b128(
          (v4i*)(gbase + (size_t)k * HT + tq),
          (v4i*)&As[k * 16 + tq], 0, 0);
    }
    __builtin_amdgcn_s_wait_asynccnt(0);
#else
    for (int i = tid; i < KDIM * 16; i += 256) {
      const int k = i >> 4, tq = i & 15;
      As[i] = gbase[(size_t)k * HT + tq];
    }
#endif
  } else {
    // fused 1x3 depthwise (pad (0,1)); clamped addresses + zero masks keep the
    // loads unconditional (no EXEC-masked branches).
    for (int i = tid; i < KDIM * 16; i += 256) {
      const int k  = i >> 4;
      const int tq = i & 15;
      const int t  = t0 + tq;
      const float* r = gbase + (size_t)k * HT;
      const int   tl = (t > 0)      ? tq - 1 : tq;
      const int   tr = (t < TT - 1) ? tq + 1 : tq;
      const float zl = (t > 0)      ? 1.0f : 0.0f;
      const float zr = (t < TT - 1) ? 1.0f : 0.0f;
      As[i] = dw3[k * 3 + 0] * (r[tl] * zl)
            + dw3[k * 3 + 1] *  r[tq]
            + dw3[k * 3 + 2] * (r[tr] * zr);
    }
  }
  __syncthreads();

  // -------- WMMA K-loop (fp32 16x16x4, native CDNA5 fp32 matrix op) --------
  const int l15  = lane & 15;
  const int hi   = lane >> 4;
  const int mrow = wv * 16 + l15;              // A row: out channel

  v8f acc = {};
#pragma unroll
  for (int k0 = 0; k0 < KDIM; k0 += 4) {
    const int k = k0 + 2 * hi;                 // per ISA 16x4 fp32 operand layout
    v2f av, bv;
    av.x = w[(size_t)mrow * KDIM + k];
    av.y = w[(size_t)mrow * KDIM + k + 1];
    bv.x = As[k * 16 + l15];                   // conflict-free: 16 consecutive dwords
    bv.y = As[(k + 1) * 16 + l15];
    acc = __builtin_amdgcn_wmma_f32_16x16x4_f32(false, av, false, bv,
                                                (short)0, acc, false, false);
  }

  // -------- epilogue + coalesced store --------
#pragma unroll
  for (int v = 0; v < 8; ++v) {
    const int ch = wv * 16 + v + 8 * hi;       // D: M = v + 8*(lane>=16)
    float r = acc[v];
    if (epi != 0) {
      const float g  = bn[ch];
      const float bb = bn[CC + ch];
      const float mu = bn[2 * CC + ch];
      const float vr = bn[3 * CC + ch];
      r = (r - mu) * (g * rsqrtf(vr + EPSV)) + bb;
      r = (epi == 1) ? fmaxf(r, 0.0f) : sigmoidf_(r);
    }
    out[((size_t)b * CC + ch) * HT + (size_t)h * TT + t0 + l15] = r;
  }
}

// ------------------------- 5x5 depthwise conv, 'same' ----------------------
__global__ void dw5x5_kernel(const float* __restrict__ in, const float* __restrict__ wgt,
                             float* __restrict__ out)
{
  size_t idx = (size_t)blockIdx.x * 256 + threadIdx.x;   // over NBIG
  int t = (int)(idx % TT); size_t r = idx / TT;
  int h = (int)(r % HH);   r /= HH;
  int c = (int)(r % CC);   int b = (int)(r / CC);
  const float* wp = wgt + c * 25;
  const float* ip = in + ((size_t)b * CC + c) * HT;
  float s = 0.0f;
#pragma unroll
  for (int i = 0; i < 5; ++i) {
    int hh = h + i - 2;
    if (hh < 0 || hh >= HH) continue;
#pragma unroll
    for (int j = 0; j < 5; ++j) {
      int tt = t + j - 2;
      if (tt < 0 || tt >= TT) continue;
      s += wp[i * 5 + j] * ip[(size_t)hh * TT + tt];
    }
  }
  out[idx] = s;
}

// ---------------- channel attention: avg/max pool over (H,T) ---------------
__global__ void pool_kernel(const float* __restrict__ in,
                            float* __restrict__ pavg, float* __restrict__ pmax)
{
  const int bc = blockIdx.x;                    // b*C + c
  const float* p = in + (size_t)bc * HT;
  float s = 0.0f, m = -3.4e38f;
  for (int i = threadIdx.x; i < (int)HT; i += 256) {
    float v = p[i]; s += v; m = fmaxf(m, v);
  }
  __shared__ float ss[256], sm[256];
  ss[threadIdx.x] = s; sm[threadIdx.x] = m;
  __syncthreads();
  for (int st = 128; st > 0; st >>= 1) {
    if ((int)threadIdx.x < st) {
      ss[threadIdx.x] += ss[threadIdx.x + st];
      sm[threadIdx.x]  = fmaxf(sm[threadIdx.x], sm[threadIdx.x + st]);
    }
    __syncthreads();
  }
  if (threadIdx.x == 0) { pavg[bc] = ss[0] / (float)HT; pmax[bc] = sm[0]; }
}

// a[b,c] = sigmoid( relu(avg@W1^T)@W2^T + relu(max@W1^T)@W2^T )
__global__ void chan_fc_kernel(const float* __restrict__ pavg, const float* __restrict__ pmax,
                               const float* __restrict__ w1, const float* __restrict__ w2,
                               float* __restrict__ a)
{
  const int b = blockIdx.x, c = threadIdx.x;
  __shared__ float hsum[MID];
  if (c < MID) {
    float ha = 0.0f, hm = 0.0f;
    for (int k = 0; k < CC; ++k) {
      float wv = w1[c * CC + k];
      ha += wv * pavg[b * CC + k];
      hm += wv * pmax[b * CC + k];
    }
    hsum[c] = fmaxf(ha, 0.0f) + fmaxf(hm, 0.0f);
  }
  __syncthreads();
  float s = 0.0f;
#pragma unroll
  for (int m = 0; m < MID; ++m) s += w2[c * MID + m] * hsum[m];
  a[b * CC + c] = sigmoidf_(s);
}

// cb0 = b1 * (b1 * a)  (outer multiply kept, faithful to source)
__global__ void apply_chan_kernel(const float* __restrict__ b1, const float* __restrict__ a,
                                  float* __restrict__ out)
{
  size_t idx = (size_t)blockIdx.x * 256 + threadIdx.x;
  size_t bc = idx / HT;
  float v = b1[idx];
  out[idx] = v * v * a[bc];
}

// ---------------- spatial attention ----------------------------------------
__global__ void spat_reduce_kernel(const float* __restrict__ x, float* __restrict__ s)
{
  size_t i = (size_t)blockIdx.x * 256 + threadIdx.x;   // over B*H*T
  int b = (int)(i / HT); size_t ht = i % HT;
  const float* p = x + (size_t)b * CC * HT + ht;
  float mean = 0.0f, mx = -3.4e38f;
  for (int c = 0; c < CC; ++c) {
    float v = p[(size_t)c * HT]; mean += v; mx = fmaxf(mx, v);
  }
  s[((size_t)b * 2 + 0) * HT + ht] = mean / (float)CC;
  s[((size_t)b * 2 + 1) * HT + ht] = mx;
}

__global__ void spat_conv_kernel(const float* __restrict__ s, const float* __restrict__ w,
                                 float* __restrict__ a)
{
  size_t i = (size_t)blockIdx.x * 256 + threadIdx.x;   // over B*H*T
  int b = (int)(i / HT); size_t ht = i % HT;
  int h = (int)(ht / TT), t = (int)(ht % TT);
  float acc = 0.0f;
  for (int ch = 0; ch < 2; ++ch)
    for (int ii = 0; ii < 7; ++ii) {
      int hh = h + ii - 3;
      if (hh < 0 || hh >= HH) continue;
      for (int jj = 0; jj < 7; ++jj) {
        int tt = t + jj - 3;
        if (tt < 0 || tt >= TT) continue;
        acc += w[(ch * 7 + ii) * 7 + jj] * s[((size_t)b * 2 + ch) * HT + (size_t)hh * TT + tt];
      }
    }
  a[i] = sigmoidf_(acc);
}

// sb = sigmoid(b2 * (b2 * a_sp))   (in-place on b2 buffer)
__global__ void spat_apply_kernel(float* __restrict__ b2, const float* __restrict__ asp)
{
  size_t idx = (size_t)blockIdx.x * 256 + threadIdx.x;
  int b = (int)(idx / ((size_t)CC * HT));
  size_t ht = idx % HT;
  float v = b2[idx];
  b2[idx] = sigmoidf_(v * v * asp[(size_t)b * HT + ht]);
}

// ---------------- SSM: input-independent scan, states S[T, SD] -------------
__global__ void ssm_scan_kernel(const float* __restrict__ A, const float* __restrict__ Bv,
                                float* __restrict__ S)
{
  const int i = threadIdx.x;                  // one wave, lane i owns s[i]
  __shared__ float sp[SD];
  float arow[SD];
#pragma unroll
  for (int j = 0; j < SD; ++j) arow[j] = A[i * SD + j];
  const float bi = Bv[i];
  float s = 0.0f;
  for (int t = 0; t < TT; ++t) {
    sp[i] = s;
    __syncthreads();
    float ns = bi;
#pragma unroll
    for (int j = 0; j < SD; ++j) ns += arow[j] * sp[j];
    s = ns;
    S[t * SD + i] = s;
    __syncthreads();
  }
}

// sc[c,t] = sum_j S[t,j] * Cm[j,c]
__global__ void sc_kernel(const float* __restrict__ S, const float* __restrict__ Cm,
                          float* __restrict__ sc)
{
  size_t i = (size_t)blockIdx.x * 256 + threadIdx.x;   // over C*T
  int c = (int)(i / TT), t = (int)(i % TT);
  float s = 0.0f;
#pragma unroll
  for (int j = 0; j < SD; ++j) s += S[t * SD + j] * Cm[j * CC + c];
  sc[(size_t)c * TT + t] = s;
}

// out = LN_c( cb + gelu(sc + cb*D) ) * sb     (LN over channel dim)
__global__ void ssm_final_kernel(const float* __restrict__ cb, const float* __restrict__ sc,
                                 const float* __restrict__ Dv, const float* __restrict__ ln,
                                 const float* __restrict__ sb, float* __restrict__ out)
{
  size_t i = (size_t)blockIdx.x * 256 + threadIdx.x;   // over B*H*T
  int b = (int)(i / HT); size_t ht = i % HT;
  int t = (int)(ht % TT);
  const size_t base = (size_t)b * CC * HT + ht;

  float mu = 0.0f, m2 = 0.0f;
  for (int c = 0; c < CC; ++c) {
    float x = cb[base + (size_t)c * HT];
    float u = x + geluf_(sc[(size_t)c * TT + t] + x * Dv[c]);
    mu += u; m2 += u * u;
  }
  mu /= (float)CC;
  float var = m2 / (float)CC - mu * mu;
  float inv = rsqrtf(var + EPSV);
  for (int c = 0; c < CC; ++c) {
    float x = cb[base + (size_t)c * HT];
    float u = x + geluf_(sc[(size_t)c * TT + t] + x * Dv[c]);
    float o = (u - mu) * inv * ln[c] + ln[CC + c];
    out[base + (size_t)c * HT] = o * sb[base + (size_t)c * HT];
  }
}

// ---------------- flip along T, in place -----------------------------------
__global__ void flipT_kernel(float* __restrict__ p)
{
  size_t idx = (size_t)blockIdx.x * 256 + threadIdx.x;  // over NBIG/2
  size_t row = idx / (TT / 2);
  int t = (int)(idx % (TT / 2));
  float* r = p + row * TT;
  float a = r[t], b = r[TT - 1 - t];
  r[t] = b; r[TT - 1 - t] = a;
}

// ===========================================================================
extern "C" void kernel_launch(void* const* d_in, const int* in_sizes, int n_in,
                              void* d_out, int out_size, void* d_ws, size_t ws_size,
                              hipStream_t stream)
{
  (void)in_sizes; (void)n_in; (void)out_size; (void)ws_size;

  const float* x     = (const float*)d_in[0];
  const float* w_in  = (const float*)d_in[1];
  const float* w_out = (const float*)d_in[2];
  const float* bn_o  = (const float*)d_in[3];

  float* ws  = (float*)d_ws;
  float* X0  = ws;                 // block input / output (flipped in place)
  float* T0  = ws + NBIG;          // dw-conv scratch / cb0
  float* B1  = ws + 2 * NBIG;      // b1 -> cb
  float* B2  = ws + 3 * NBIG;      // b2 -> sb
  float* Sm  = ws + 4 * NBIG;                       // [T, SD]
  float* SC  = Sm + (size_t)TT * SD;                // [C, T]
  float* PA  = SC + (size_t)CC * TT;                // [B*C]
  float* PM  = PA + BB * CC;                        // [B*C]
  float* AC  = PM + BB * CC;                        // [B*C]
  float* SSP = AC + BB * CC;                        // [B,2,H,T]
  float* ASP = SSP + (size_t)BB * 2 * HT;           // [B,H,T]

  const int GEMM_BLOCKS = BB * HH * (TT / 16);      // 16384 tiles, 8 waves each
  const int ELT_BLOCKS  = (int)(NBIG / 256);        // 131072
  const int BHT_BLOCKS  = (int)((size_t)BB * HT / 256); // 1024

  auto run_block = [&](const float* inp, float* outp, void* const* P) {
    const float* dwa = (const float*)P[0];  const float* pwa = (const float*)P[1];
    const float* bna = (const float*)P[2];  const float* dwb = (const float*)P[3];
    const float* pwb = (const float*)P[4];  const float* bnb = (const float*)P[5];
    const float* ca1 = (const float*)P[6];  const float* ca2 = (const float*)P[7];
    const float* saw = (const float*)P[8];  const float* d1w = (const float*)P[9];
    const float* d1p = (const float*)P[10]; const float* d1b = (const float*)P[11];
    const float* Am  = (const float*)P[12]; const float* Bv  = (const float*)P[13];
    const float* Cm  = (const float*)P[14]; const float* Dv  = (const float*)P[15];
    const float* lnp = (const float*)P[16];

    dw5x5_kernel<<<ELT_BLOCKS, 256, 0, stream>>>(inp, dwa, T0);
    pw_gemm_kernel<CC><<<GEMM_BLOCKS, 256, 0, stream>>>(T0, pwa, bna, nullptr, B1, 1); // b1
    dw5x5_kernel<<<ELT_BLOCKS, 256, 0, stream>>>(inp, dwb, T0);
    pw_gemm_kernel<CC><<<GEMM_BLOCKS, 256, 0, stream>>>(T0, pwb, bnb, nullptr, B2, 1); // b2

    pool_kernel<<<BB * CC, 256, 0, stream>>>(B1, PA, PM);
    chan_fc_kernel<<<BB, CC, 0, stream>>>(PA, PM, ca1, ca2, AC);
    apply_chan_kernel<<<ELT_BLOCKS, 256, 0, stream>>>(B1, AC, T0);                     // cb0 -> T0
    // fused 1x3 depthwise + pointwise + BN + sigmoid:  cb -> B1
    pw_gemm_kernel<CC><<<GEMM_BLOCKS, 256, 0, stream>>>(T0, d1p, d1b, d1w, B1, 2);

    spat_reduce_kernel<<<BHT_BLOCKS, 256, 0, stream>>>(B2, SSP);
    spat_conv_kernel<<<BHT_BLOCKS, 256, 0, stream>>>(SSP, saw, ASP);
    spat_apply_kernel<<<ELT_BLOCKS, 256, 0, stream>>>(B2, ASP);                        // sb -> B2

    ssm_scan_kernel<<<1, SD, 0, stream>>>(Am, Bv, Sm);
    sc_kernel<<<(int)((size_t)CC * TT / 256), 256, 0, stream>>>(Sm, Cm, SC);
    ssm_final_kernel<<<BHT_BLOCKS, 256, 0, stream>>>(B1, SC, Dv, lnp, B2, outp);
  };

  // stem: 1x1 conv 64 -> 128 (WMMA GEMM)
  pw_gemm_kernel<CIN><<<GEMM_BLOCKS, 256, 0, stream>>>(x, w_in, nullptr, nullptr, X0, 0);

  run_block(X0, X0, d_in + 4);                                   // block 1
  flipT_kernel<<<(int)(NBIG / 2 / 256), 256, 0, stream>>>(X0);
  run_block(X0, X0, d_in + 21);                                  // block 2
  flipT_kernel<<<(int)(NBIG / 2 / 256), 256, 0, stream>>>(X0);

  // head: 1x1 conv 128 -> 128 + BN + ReLU -> d_out
  pw_gemm_kernel<CC><<<GEMM_BLOCKS, 256, 0, stream>>>(X0, w_out, bn_o, nullptr,
                                                      (float*)d_out, 1);
}